// BaseLSSFPN_80522046866052
// MI455X (gfx1250) — compile-verified
//
#include <hip/hip_runtime.h>
#include <hip/hip_bf16.h>

// ---------------------------------------------------------------------------
// BaseLSSFPN (lift-splat-shoot BEV pooling) for MI455X / gfx1250.
// B=2, N=6, D=112, C=80, FH=16, FW=44, VX=VY=128, VZ=1.
// Points = 12*112*16*44 = 946176 (divisible by 16; tiles never cross cameras).
// Output (2,80,128,128) f32 = 2,621,440 elements.
// ---------------------------------------------------------------------------

#define NPTS      946176
#define PER_BN    78848      // D*FH*FW = 112*704
#define PER_D     704        // FH*FW
#define FW_       44
#define FH_       16
#define D_        112
#define C_        80
#define NCH       192        // D + C
#define OUT_N     2621440    // 2*80*128*128
#define OUT_BSTR  1310720    // 80*128*128 (per-batch stride at c=0)
#define OUT_CSTR  16384      // 128*128

typedef __attribute__((ext_vector_type(2))) float v2f;
typedef __attribute__((ext_vector_type(8))) float v8f;

// ------------------------- 4x4 helpers (setup) ------------------------------
__device__ inline void inv4(const float* m, float* o) {
    float inv[16];
    inv[0]  =  m[5]*m[10]*m[15]-m[5]*m[11]*m[14]-m[9]*m[6]*m[15]+m[9]*m[7]*m[14]+m[13]*m[6]*m[11]-m[13]*m[7]*m[10];
    inv[4]  = -m[4]*m[10]*m[15]+m[4]*m[11]*m[14]+m[8]*m[6]*m[15]-m[8]*m[7]*m[14]-m[12]*m[6]*m[11]+m[12]*m[7]*m[10];
    inv[8]  =  m[4]*m[9]*m[15]-m[4]*m[11]*m[13]-m[8]*m[5]*m[15]+m[8]*m[7]*m[13]+m[12]*m[5]*m[11]-m[12]*m[7]*m[9];
    inv[12] = -m[4]*m[9]*m[14]+m[4]*m[10]*m[13]+m[8]*m[5]*m[14]-m[8]*m[6]*m[13]-m[12]*m[5]*m[10]+m[12]*m[6]*m[9];
    inv[1]  = -m[1]*m[10]*m[15]+m[1]*m[11]*m[14]+m[9]*m[2]*m[15]-m[9]*m[3]*m[14]-m[13]*m[2]*m[11]+m[13]*m[3]*m[10];
    inv[5]  =  m[0]*m[10]*m[15]-m[0]*m[11]*m[14]-m[8]*m[2]*m[15]+m[8]*m[3]*m[14]+m[12]*m[2]*m[11]-m[12]*m[3]*m[10];
    inv[9]  = -m[0]*m[9]*m[15]+m[0]*m[11]*m[13]+m[8]*m[1]*m[15]-m[8]*m[3]*m[13]-m[12]*m[1]*m[11]+m[12]*m[3]*m[9];
    inv[13] =  m[0]*m[9]*m[14]-m[0]*m[10]*m[13]-m[8]*m[1]*m[14]+m[8]*m[2]*m[13]+m[12]*m[1]*m[10]-m[12]*m[2]*m[9];
    inv[2]  =  m[1]*m[6]*m[15]-m[1]*m[7]*m[14]-m[5]*m[2]*m[15]+m[5]*m[3]*m[14]+m[13]*m[2]*m[7]-m[13]*m[3]*m[6];
    inv[6]  = -m[0]*m[6]*m[15]+m[0]*m[7]*m[14]+m[4]*m[2]*m[15]-m[4]*m[3]*m[14]-m[12]*m[2]*m[7]+m[12]*m[3]*m[6];
    inv[10] =  m[0]*m[5]*m[15]-m[0]*m[7]*m[13]-m[4]*m[1]*m[15]+m[4]*m[3]*m[13]+m[12]*m[1]*m[7]-m[12]*m[3]*m[5];
    inv[14] = -m[0]*m[5]*m[14]+m[0]*m[6]*m[13]+m[4]*m[1]*m[14]-m[4]*m[2]*m[13]-m[12]*m[1]*m[6]+m[12]*m[2]*m[5];
    inv[3]  = -m[1]*m[6]*m[11]+m[1]*m[7]*m[10]+m[5]*m[2]*m[11]-m[5]*m[3]*m[10]-m[9]*m[2]*m[7]+m[9]*m[3]*m[6];
    inv[7]  =  m[0]*m[6]*m[11]-m[0]*m[7]*m[10]-m[4]*m[2]*m[11]+m[4]*m[3]*m[10]+m[8]*m[2]*m[7]-m[8]*m[3]*m[6];
    inv[11] = -m[0]*m[5]*m[11]+m[0]*m[7]*m[9]+m[4]*m[1]*m[11]-m[4]*m[3]*m[9]-m[8]*m[1]*m[7]+m[8]*m[3]*m[5];
    inv[15] =  m[0]*m[5]*m[10]-m[0]*m[6]*m[9]-m[4]*m[1]*m[10]+m[4]*m[2]*m[9]+m[8]*m[1]*m[6]-m[8]*m[2]*m[5];
    float det = m[0]*inv[0] + m[1]*inv[4] + m[2]*inv[8] + m[3]*inv[12];
    det = 1.0f / det;
    for (int i = 0; i < 16; ++i) o[i] = inv[i] * det;
}

__device__ inline void mm4(const float* a, const float* b, float* c) {
    for (int i = 0; i < 4; ++i)
        for (int j = 0; j < 4; ++j) {
            float s = 0.0f;
            for (int k = 0; k < 4; ++k) s += a[i*4+k] * b[k*4+j];
            c[i*4+j] = s;
        }
}

// Per (b,n): idaInv = inv(ida); Mfull = bda @ s2e @ inv(intrin). 12 threads.
__global__ void setup_mats(const float* __restrict__ s2e,
                           const float* __restrict__ intrin,
                           const float* __restrict__ ida,
                           const float* __restrict__ bda,
                           float* __restrict__ idaInv,
                           float* __restrict__ Mfull) {
    int t = threadIdx.x;
    if (t >= 12) return;
    int b = t / 6;
    float tmp[16], t1[16], t2[16];
    inv4(ida + t*16, tmp);
    for (int i = 0; i < 16; ++i) idaInv[t*16 + i] = tmp[i];
    inv4(intrin + t*16, tmp);          // intrin^-1
    mm4(s2e + t*16, tmp, t1);          // s2e @ intrin^-1
    mm4(bda + b*16, t1, t2);           // bda @ (..)
    for (int i = 0; i < 16; ++i) Mfull[t*16 + i] = t2[i];
}

// ------------------------- geometry via WMMA --------------------------------
// Each wave transforms 16 points with two V_WMMA_F32_16X16X4_F32 ops.
// A-layout (16x4 f32): lanes 0-15 hold {K0,K1}, lanes 16-31 hold {K2,K3}, M=lane%16.
// B (4x16) = matrix^T padded with zero columns beyond col 3.
// C/D layout: lane = column N (0-15 twice), VGPR r = row M=r (+8 for upper lanes).
__global__ void __launch_bounds__(256)
geom_wmma(const float* __restrict__ idaInv,
          const float* __restrict__ Mfull,
          int* __restrict__ vox) {
    __shared__ float ldsP[8][16][4];
    const int lane = threadIdx.x & 31;
    const int wl   = threadIdx.x >> 5;
    const int m    = lane & 15;         // point within tile / B column
    const int hf   = lane >> 4;         // 0: K0/K1 half, 1: K2/K3 half
    const int tile = blockIdx.x * 8 + wl;
    const int pt   = tile * 16 + m;

    // decompose point index -> (bn, d, h, w); bn is uniform across the tile
    const int bn = pt / PER_BN;
    int r1 = pt - bn * PER_BN;
    const int dd = r1 / PER_D;
    int r2 = r1 - dd * PER_D;
    const int hh = r2 / FW_;
    const int ww = r2 - hh * FW_;

    // frustum coordinates (compile-time grid)
    const float fx = (float)ww * (703.0f / 43.0f);   // linspace(0,703,44)
    const float fy = (float)hh * 17.0f;              // linspace(0,255,16)
    const float fd = 2.0f + 0.5f * (float)dd;

    v2f a;
    if (hf == 0) { a.x = fx; a.y = fy; } else { a.x = fd; a.y = 1.0f; }

    const float* I = idaInv + bn * 16;
    v2f bm;
    if (m < 4) { bm.x = I[m*4 + (hf ? 2 : 0)]; bm.y = I[m*4 + (hf ? 3 : 1)]; }
    else       { bm.x = 0.0f; bm.y = 0.0f; }

    v8f cz = {0.f,0.f,0.f,0.f,0.f,0.f,0.f,0.f};
    v8f d1 = __builtin_amdgcn_wmma_f32_16x16x4_f32(false, a, false, bm,
                                                   (short)0, cz, false, false);
    if (m < 4) {
        #pragma unroll
        for (int r = 0; r < 8; ++r) ldsP[wl][r + hf*8][m] = d1[r];
    }
    __syncthreads();

    // perspective step: (x,y,z,w) -> (x*z, y*z, z, w)
    const float px = ldsP[wl][m][0];
    const float py = ldsP[wl][m][1];
    const float pz = ldsP[wl][m][2];
    const float pw = ldsP[wl][m][3];
    v2f a2;
    if (hf == 0) { a2.x = px * pz; a2.y = py * pz; } else { a2.x = pz; a2.y = pw; }

    const float* MM = Mfull + bn * 16;
    v2f b2;
    if (m < 4) { b2.x = MM[m*4 + (hf ? 2 : 0)]; b2.y = MM[m*4 + (hf ? 3 : 1)]; }
    else       { b2.x = 0.0f; b2.y = 0.0f; }
    __syncthreads();   // all reads of ldsP done before overwrite

    v8f d2 = __builtin_amdgcn_wmma_f32_16x16x4_f32(false, a2, false, b2,
                                                   (short)0, cz, false, false);
    if (m < 4) {
        #pragma unroll
        for (int r = 0; r < 8; ++r) ldsP[wl][r + hf*8][m] = d2[r];
    }
    __syncthreads();

    if (hf == 0) {
        const float gx = ldsP[wl][m][0];
        const float gy = ldsP[wl][m][1];
        const float gz = ldsP[wl][m][2];
        // reference: int32 cast truncates toward zero, then bounds check
        const int ix = (int)((gx + 51.2f) / 0.8f);
        const int iy = (int)((gy + 51.2f) / 0.8f);
        const int iz = (int)((gz + 5.0f) / 8.0f);
        const bool valid = (ix >= 0) & (ix < 128) & (iy >= 0) & (iy < 128) &
                           (iz >= 0) & (iz < 1);
        const int b = bn / 6;
        vox[pt] = valid ? (b * OUT_BSTR + iy * 128 + ix) : -1;
    }
}

// ------------------------- zero the BEV output ------------------------------
__global__ void zero_out(float* __restrict__ out, int n) {
    int i = blockIdx.x * blockDim.x + threadIdx.x;
    if (i < n) out[i] = 0.0f;
}

// ------------------------- softmax + lift + scatter -------------------------
// One block per pixel (bn,h,w). 128 threads (4 waves).
__global__ void __launch_bounds__(128)
scatter_bev(const float* __restrict__ df,
            const int* __restrict__ vox,
            float* __restrict__ out) {
    __shared__ float sw[D_];      // softmax weights
    __shared__ float sfeat[C_];   // 80 feature channels
    __shared__ float red[128];    // reduction buffer

    const int pix = blockIdx.x;
    const int bn  = pix / PER_D;
    int r = pix - bn * PER_D;
    const int hh = r / FW_;
    const int ww = r - hh * FW_;
    const int t  = threadIdx.x;

    const float* base = df + (long)bn * (NCH * PER_D) + hh * FW_ + ww;

    // ---- block softmax over 112 depth logits ----
    float l = -3.4e38f;
    if (t < D_) l = base[t * PER_D];
    red[t] = l;
    __syncthreads();
    #pragma unroll
    for (int s = 64; s > 0; s >>= 1) {
        if (t < s) red[t] = fmaxf(red[t], red[t + s]);
        __syncthreads();
    }
    const float mx = red[0];
    __syncthreads();
    const float e = (t < D_) ? __expf(l - mx) : 0.0f;
    red[t] = e;
    __syncthreads();
    #pragma unroll
    for (int s = 64; s > 0; s >>= 1) {
        if (t < s) red[t] += red[t + s];
        __syncthreads();
    }
    const float denom = red[0];
    if (t < D_) sw[t] = e / denom;

    // ---- stage 80 feature channels into LDS via async-tensor path ----
    if (t < C_) {
        unsigned loff = (unsigned)(size_t)&sfeat[t];
        const float* g = base + (D_ + t) * PER_D;
        asm volatile("global_load_async_to_lds_b32 %0, %1, off"
                     :: "v"(loff), "v"((unsigned long long)(size_t)g)
                     : "memory");
    }
    asm volatile("s_wait_asynccnt 0x0" ::: "memory");
    __syncthreads();

    // ---- scatter: 112 depth bins x 80 channels of no-return f32 atomics ----
    if (t < D_) {
        const int vi = vox[((bn * D_ + t) * FH_ + hh) * FW_ + ww];
        if (vi >= 0) {
            const float wt = sw[t];
            #pragma unroll 8
            for (int c = 0; c < C_; ++c)
                atomicAdd(out + vi + c * OUT_CSTR, wt * sfeat[c]);
        }
    }
}

// ---------------------------------------------------------------------------
extern "C" void kernel_launch(void* const* d_in, const int* in_sizes, int n_in,
                              void* d_out, int out_size, void* d_ws, size_t ws_size,
                              hipStream_t stream) {
    const float* df     = (const float*)d_in[0];  // (12,192,16,44)
    const float* s2e    = (const float*)d_in[1];  // (2,6,4,4)
    const float* intrin = (const float*)d_in[2];  // (2,6,4,4)
    const float* ida    = (const float*)d_in[3];  // (2,6,4,4)
    const float* bda    = (const float*)d_in[4];  // (2,4,4)
    float* out = (float*)d_out;

    // workspace: [0..191] idaInv, [192..383] Mfull, then 946176 int voxel ids
    float* wsf    = (float*)d_ws;
    float* idaInv = wsf;
    float* Mfull  = wsf + 192;
    int*   vox    = (int*)(wsf + 384);

    setup_mats<<<1, 16, 0, stream>>>(s2e, intrin, ida, bda, idaInv, Mfull);

    // 946176 points / 16 per wave = 59136 waves; 8 waves per 256-thread block
    geom_wmma<<<59136 / 8, 256, 0, stream>>>(idaInv, Mfull, vox);

    zero_out<<<(OUT_N + 255) / 256, 256, 0, stream>>>(out, OUT_N);

    // one block per pixel: 12 * 16 * 44 = 8448
    scatter_bev<<<8448, 128, 0, stream>>>(df, vox, out);
}